// GATLayerST_59115929862466
// MI455X (gfx1250) — compile-verified
//
#include <hip/hip_runtime.h>

typedef __attribute__((ext_vector_type(2))) float v2f;
typedef __attribute__((ext_vector_type(8))) float v8f;

#define IN_F  128
#define OUT_F 128

// ---------------------------------------------------------------------------
// Zero-fill output (harness poisons d_out; atomics need a zero base).
// ---------------------------------------------------------------------------
__global__ void zero_f4(float4* __restrict__ p, int n4) {
    int i = blockIdx.x * blockDim.x + threadIdx.x;
    if (i < n4) p[i] = make_float4(0.f, 0.f, 0.f, 0.f);
}

// ---------------------------------------------------------------------------
// support = x @ W  via V_WMMA_F32_16X16X4_F32.
// Block = 128 threads (4 waves). Each wave owns a 16-row M tile and the full
// N=128 (8 accumulators of v8f).
//
// W is staged into LDS in a fragment-native PAIR layout:
//   sW2[kp * 128 + n] = { W[2kp][n], W[2kp+1][n] }
// so a B fragment is a single aligned ds_load_b64 landing in an even VGPR
// pair (no v_mov shuffles before the WMMA).
//
// A fragment (16x4 f32): lanes 0-15 hold K=k0,k0+1; lanes 16-31 hold
// K=k0+2,k0+3  -> single global_load_b64 per lane per K-step.
// C/D (16x16 f32): acc[n][i] -> row m0 + i + 8*(lane>=16), col n*16 + lane%16.
// ---------------------------------------------------------------------------
__global__ __launch_bounds__(128) void gemm_wmma(const float* __restrict__ x,
                                                 const float* __restrict__ W,
                                                 float* __restrict__ support,
                                                 int n_nodes) {
    __shared__ v2f sW2[(IN_F / 2) * OUT_F];   // 64 pairs x 128 cols = 64 KB

    // Stage W into pair layout: sW2[kp*128 + n] = {W[2kp][n], W[2kp+1][n]}
    for (int idx = threadIdx.x; idx < (IN_F / 2) * OUT_F; idx += 128) {
        int kp = idx >> 7;          // pair index 0..63
        int n  = idx & 127;         // column
        v2f p;
        p.x = W[(2 * kp)     * OUT_F + n];
        p.y = W[(2 * kp + 1) * OUT_F + n];
        sW2[idx] = p;
    }
    __syncthreads();

    const int wave = threadIdx.x >> 5;
    const int lane = threadIdx.x & 31;
    const int m0   = (blockIdx.x * 4 + wave) * 16;
    if (m0 >= n_nodes) return;                    // wave-uniform (n_nodes % 16 == 0)

    const int halfsel = lane >> 4;                // 0 or 1
    const int col     = lane & 15;
    const int kofs    = halfsel * 2;              // K sub-offset for A/B fragments

    const float* xrow = x + (size_t)(m0 + col) * IN_F;

    v8f acc[8] = {};

    for (int k0 = 0; k0 < IN_F; k0 += 4) {
        v2f a = *(const v2f*)(xrow + k0 + kofs);          // global_load_b64
        const v2f* brow = sW2 + ((k0 >> 1) + halfsel) * OUT_F;
        #pragma unroll
        for (int n = 0; n < 8; ++n) {
            v2f b = brow[n * 16 + col];                   // ds_load_b64
            acc[n] = __builtin_amdgcn_wmma_f32_16x16x4_f32(
                /*neg_a=*/false, a, /*neg_b=*/false, b,
                /*c_mod=*/(short)0, acc[n],
                /*reuse_a=*/false, /*reuse_b=*/false);
        }
    }

    #pragma unroll
    for (int n = 0; n < 8; ++n) {
        #pragma unroll
        for (int i = 0; i < 8; ++i) {
            int row = m0 + i + halfsel * 8;
            support[(size_t)row * OUT_F + n * 16 + col] = acc[n][i];
        }
    }
}

// ---------------------------------------------------------------------------
// out[row[e]] += w[e] * support[col[e]]   (one wave32 per edge)
// Lane l handles feats [4l, 4l+3]: one b128 gather (L2-resident, 512 B/wave
// coalesced) + 4 native f32 atomic adds to contiguous addresses.
// ---------------------------------------------------------------------------
__global__ __launch_bounds__(256) void spmm_scatter(const long long* __restrict__ ei,
                                                    const float* __restrict__ ew,
                                                    const float* __restrict__ support,
                                                    float* __restrict__ out,
                                                    int n_edges) {
    int gwave = (blockIdx.x * blockDim.x + threadIdx.x) >> 5;
    int lane  = threadIdx.x & 31;
    if (gwave >= n_edges) return;                 // wave-uniform

    long long r = ei[gwave];                      // destination row
    long long c = ei[(size_t)n_edges + gwave];    // source column
    float     w = ew[gwave];

    const float4 v = ((const float4*)(support + (size_t)c * OUT_F))[lane];
    float* dst = out + (size_t)r * OUT_F + lane * 4;

    unsafeAtomicAdd(dst + 0, v.x * w);            // global_atomic_add_f32 (no ret)
    unsafeAtomicAdd(dst + 1, v.y * w);
    unsafeAtomicAdd(dst + 2, v.z * w);
    unsafeAtomicAdd(dst + 3, v.w * w);
}

// ---------------------------------------------------------------------------
extern "C" void kernel_launch(void* const* d_in, const int* in_sizes, int n_in,
                              void* d_out, int out_size, void* d_ws, size_t ws_size,
                              hipStream_t stream) {
    const float*     x  = (const float*)d_in[0];      // [N, 128] f32
    const long long* ei = (const long long*)d_in[1];  // [2, E]  i64
    const float*     ew = (const float*)d_in[2];      // [E]     f32
    const float*     W  = (const float*)d_in[3];      // [128, 128] f32
    float* out      = (float*)d_out;                  // [N, 128] f32
    float* support  = (float*)d_ws;                   // [N, 128] f32 scratch

    const int n_nodes = in_sizes[0] / IN_F;
    const int n_edges = in_sizes[2];

    // 1) zero the output
    int n4 = out_size / 4;
    zero_f4<<<(n4 + 255) / 256, 256, 0, stream>>>((float4*)out, n4);

    // 2) support = x @ W  (WMMA). 4 waves/block, 16 rows/wave -> 64 rows/block
    int mtiles = (n_nodes + 15) / 16;
    int blocks = (mtiles + 3) / 4;
    gemm_wmma<<<blocks, 128, 0, stream>>>(x, W, support, n_nodes);

    // 3) edge scatter: one wave per edge
    long long threads = (long long)n_edges * 32;
    int sblocks = (int)((threads + 255) / 256);
    spmm_scatter<<<sblocks, 256, 0, stream>>>(ei, ew, support, out, n_edges);
}